// bilinear_interpolate_47639777247674
// MI455X (gfx1250) — compile-verified
//
#include <hip/hip_runtime.h>

// Problem constants (from reference)
#define RH 64
#define RW 64
#define RPH 7
#define RPW 7
#define RGH 2
#define RGW 2
#define NROI 16384
#define SPR (RPH * RPW * RGH * RGW)   // 196 samples per ROI
#define TOTAL (NROI * SPR)            // 3,211,264 samples (divisible by 256)

typedef float v4f __attribute__((ext_vector_type(4)));

__global__ __launch_bounds__(256) void roi_bilinear_coef_kernel(
    const float* __restrict__ roi_start_h,
    const float* __restrict__ roi_start_w,
    const float* __restrict__ bin_size_h,
    const float* __restrict__ bin_size_w,
    float* __restrict__ out)
{
    const unsigned idx = blockIdx.x * 256u + threadIdx.x;   // sample index, < TOTAL exactly

    // Decompose: idx = n*196 + ph*28 + pw*4 + gy*2 + gx
    const unsigned n  = idx / SPR;                // magic-multiply, no HW div
    const unsigned s  = idx - n * SPR;
    const unsigned ph = s / (RPW * RGH * RGW);    // s / 28
    const unsigned r  = s - ph * (RPW * RGH * RGW);
    const unsigned pw = r >> 2;                   // r / 4
    const unsigned gy = (r >> 1) & 1u;
    const unsigned gx = r & 1u;

    // Per-ROI params (wave spans <=2 ROIs -> L0 hits)
    const float sh = roi_start_h[n];
    const float sw = roi_start_w[n];
    const float bh = bin_size_h[n];
    const float bw = bin_size_w[n];

    // y = start + p*b + ((g+0.5)/G)*b   (matches reference eval order)
    const float y = sh + (float)ph * bh + ((float)gy + 0.5f) * 0.5f * bh;
    const float x = sw + (float)pw * bw + ((float)gx + 0.5f) * 0.5f * bw;

    const bool oob = (y < -1.0f) | (y > (float)RH) | (x < -1.0f) | (x > (float)RW);

    // _low_high for y (size 64)
    float c0   = fmaxf(y, 0.0f);
    float low0 = floorf(c0);
    bool  edgy = low0 >= (float)(RH - 1);
    float y_low  = edgy ? (float)(RH - 1) : low0;
    float y_high = edgy ? (float)(RH - 1) : low0 + 1.0f;
    float ly     = edgy ? 0.0f            : (c0 - low0);
    float hy     = 1.0f - ly;

    // _low_high for x (size 64)
    c0   = fmaxf(x, 0.0f);
    low0 = floorf(c0);
    bool edgx = low0 >= (float)(RW - 1);
    float x_low  = edgx ? (float)(RW - 1) : low0;
    float x_high = edgx ? (float)(RW - 1) : low0 + 1.0f;
    float lx     = edgx ? 0.0f            : (c0 - low0);
    float hx     = 1.0f - lx;

    const float ylW = y_low  * (float)RW;
    const float yhW = y_high * (float)RW;

    v4f pos, wgt;
    pos.x = ylW + x_low;
    pos.y = ylW + x_high;
    pos.z = yhW + x_low;
    pos.w = yhW + x_high;
    wgt.x = hy * hx;
    wgt.y = hy * lx;
    wgt.z = ly * hx;
    wgt.w = ly * lx;

    if (oob) {
        pos = (v4f){0.f, 0.f, 0.f, 0.f};
        wgt = (v4f){0.f, 0.f, 0.f, 0.f};
    }

    // 8 contiguous floats per sample -> two 16B non-temporal stores
    // (global_store_b128 th:NT). Each wave writes a contiguous 1 KB span.
    v4f* dst = reinterpret_cast<v4f*>(out) + (size_t)idx * 2u;
    __builtin_nontemporal_store(pos, dst);
    __builtin_nontemporal_store(wgt, dst + 1);
}

extern "C" void kernel_launch(void* const* d_in, const int* in_sizes, int n_in,
                              void* d_out, int out_size, void* d_ws, size_t ws_size,
                              hipStream_t stream) {
    (void)in_sizes; (void)n_in; (void)out_size; (void)d_ws; (void)ws_size;
    const float* rsh = (const float*)d_in[0];
    const float* rsw = (const float*)d_in[1];
    const float* bsh = (const float*)d_in[2];
    const float* bsw = (const float*)d_in[3];
    float* out = (float*)d_out;

    const int block = 256;                 // 8 wave32 waves
    const int grid  = TOTAL / block;       // 12544 blocks, exact
    roi_bilinear_coef_kernel<<<grid, block, 0, stream>>>(rsh, rsw, bsh, bsw, out);
}